// SelfAttentionWrapper_56839597195736
// MI455X (gfx1250) — compile-verified
//
#include <hip/hip_runtime.h>
#include <hip/hip_bf16.h>
#include <math.h>

// Problem constants (from reference): B=1
#define S_   256
#define HID_ 1024
#define H_   16
#define KV_  8
#define D_   64
static constexpr float SCALE = 0.125f; // 1/sqrt(64)

typedef __attribute__((ext_vector_type(2))) float v2f;
typedef __attribute__((ext_vector_type(8))) float v8f;

// f32 WMMA: D(16x16) = A(16x4) * B(4x16) + C. 8-arg pattern:
// (neg_a, A, neg_b, B, c_mod, C, reuse_a, reuse_b)
__device__ inline v8f wmma_k4(v2f a, v2f b, v8f c) {
  return __builtin_amdgcn_wmma_f32_16x16x4_f32(false, a, false, b, (short)0, c,
                                               false, false);
}

// ---------------------------------------------------------------------------
// Kernel 1: fused QKV projection. hidden(256x1024) @ [Wq|Wk|Wv] -> 2048 cols.
// One wave per 16x16 tile; writes to per-head [head][s][d] layout.
__global__ __launch_bounds__(32) void qkv_gemm(
    const float* __restrict__ hs, const float* __restrict__ Wq,
    const float* __restrict__ Wk, const float* __restrict__ Wv,
    float* __restrict__ q, float* __restrict__ k, float* __restrict__ v) {
  int tile = blockIdx.x;       // 0..2047
  int tm = tile & 15;          // 16 row tiles (S/16)
  int tn = tile >> 4;          // 128 col tiles (2048/16)
  int lane = threadIdx.x, half = lane >> 4, l = lane & 15;
  int m0 = tm * 16, col0 = tn * 16;

  const float* W;
  int ldb, cbase;
  if (col0 < 1024)      { W = Wq; ldb = 1024; cbase = col0; }
  else if (col0 < 1536) { W = Wk; ldb = 512;  cbase = col0 - 1024; }
  else                  { W = Wv; ldb = 512;  cbase = col0 - 1536; }

  v8f acc = {};
  const float* ap = hs + (m0 + l) * HID_ + 2 * half;   // A: row-major [s][hid]
  const float* bp = W + (2 * half) * ldb + cbase + l;  // B: row-major [hid][col]
  for (int kk = 0; kk < HID_; kk += 4) {
    v2f a = { ap[kk], ap[kk + 1] };
    v2f b = { bp[kk * ldb], bp[kk * ldb + ldb] };
    acc = wmma_k4(a, b, acc);
  }

  int gc = col0 + l;
  for (int r = 0; r < 8; ++r) {
    int row = m0 + r + 8 * half;
    float val = acc[r];
    if (gc < 1024) {
      int h = gc >> 6, d = gc & 63;
      q[(h * S_ + row) * D_ + d] = val;
    } else if (gc < 1536) {
      int c = gc - 1024, h = c >> 6, d = c & 63;
      k[(h * S_ + row) * D_ + d] = val;
    } else {
      int c = gc - 1536, h = c >> 6, d = c & 63;
      v[(h * S_ + row) * D_ + d] = val;
    }
  }
}

// ---------------------------------------------------------------------------
// Kernel 2: RoPE applied in-place to q and k. One thread per (head, s, d-pair).
__global__ __launch_bounds__(256) void rope_kernel(
    float* __restrict__ q, float* __restrict__ k,
    const float* __restrict__ cosb, const float* __restrict__ sinb) {
  int tid = blockIdx.x * blockDim.x + threadIdx.x; // (H+KV)*S*32 threads
  int d = tid & 31;
  int rest = tid >> 5;
  int s = rest & (S_ - 1);
  int head = rest >> 8;
  float* base = (head < H_) ? (q + (head * S_ + s) * D_)
                            : (k + ((head - H_) * S_ + s) * D_);
  float c1 = cosb[s * D_ + d],      s1 = sinb[s * D_ + d];
  float c2 = cosb[s * D_ + d + 32], s2 = sinb[s * D_ + d + 32];
  float x1 = base[d], x2 = base[d + 32];
  base[d]      = x1 * c1 - x2 * s1; // x*cos + (-x2)*sin
  base[d + 32] = x2 * c2 + x1 * s2; // x*cos + ( x1)*sin
}

// ---------------------------------------------------------------------------
// Kernel 3: attention scores aw[h][s][t] = scale*(q.k) + mask. WMMA, K=64.
__global__ __launch_bounds__(32) void scores_gemm(
    const float* __restrict__ q, const float* __restrict__ k,
    const float* __restrict__ mask, float* __restrict__ aw) {
  int blk = blockIdx.x;            // h*256 + tm*16 + tn
  int h = blk >> 8;
  int tm = (blk >> 4) & 15, tn = blk & 15;
  int lane = threadIdx.x, half = lane >> 4, l = lane & 15;
  int m0 = tm * 16, n0 = tn * 16;
  const float* A = q + h * S_ * D_;        // [s][d]
  const float* B = k + (h >> 1) * S_ * D_; // [t][d]: B^T access is contiguous
  v8f acc = {};
  const float* ap = A + (m0 + l) * D_ + 2 * half;
  const float* bp = B + (n0 + l) * D_ + 2 * half;
  for (int kk = 0; kk < D_; kk += 4) {
    v2f a = { ap[kk], ap[kk + 1] };
    v2f b = { bp[kk], bp[kk + 1] };
    acc = wmma_k4(a, b, acc);
  }
  for (int r = 0; r < 8; ++r) {
    int srow = m0 + r + 8 * half, tcol = n0 + l;
    aw[(h * S_ + srow) * S_ + tcol] = acc[r] * SCALE + mask[srow * S_ + tcol];
  }
}

// ---------------------------------------------------------------------------
// Kernel 4: row softmax of aw -> p. One wave32 per row of 256 (8 elems/lane).
__global__ __launch_bounds__(256) void softmax_rows(
    const float* __restrict__ aw, float* __restrict__ p) {
  int row = (blockIdx.x * blockDim.x + threadIdx.x) >> 5; // 0..4095
  int lane = threadIdx.x & 31;
  const float* r = aw + row * S_;
  float x[8], mx = -3.402823466e38f;
  for (int i = 0; i < 8; ++i) { x[i] = r[lane + 32 * i]; mx = fmaxf(mx, x[i]); }
  for (int o = 16; o >= 1; o >>= 1) mx = fmaxf(mx, __shfl_xor(mx, o, 32));
  float sum = 0.f;
  for (int i = 0; i < 8; ++i) { x[i] = __expf(x[i] - mx); sum += x[i]; }
  for (int o = 16; o >= 1; o >>= 1) sum += __shfl_xor(sum, o, 32);
  float inv = 1.f / sum;
  float* pr = p + row * S_;
  for (int i = 0; i < 8; ++i) pr[lane + 32 * i] = x[i] * inv;
}

// ---------------------------------------------------------------------------
// Kernel 5 (dominant cost): per-(h,d) diff softmax + qk_importance.
// diff[t] = (aw[h,s,t] - q[h,s,d]*k[h,t,d]) * scale; softmax over t;
// acc += (p_diff - p)^2 over all s,t. One block per (h,d), 8 waves = 8 s-lanes.
__global__ __launch_bounds__(256) void qk_imp_kernel(
    const float* __restrict__ q, const float* __restrict__ k,
    const float* __restrict__ aw, const float* __restrict__ p,
    float* __restrict__ qk_imp) {
  int h = blockIdx.x >> 6;
  int d = blockIdx.x & 63;
  int wave = threadIdx.x >> 5, lane = threadIdx.x & 31;
  const float* kh = k + (h >> 1) * S_ * D_;
  float kd[8];
  for (int i = 0; i < 8; ++i) kd[i] = kh[(lane + 32 * i) * D_ + d]; // k[:,d] cached
  float acc = 0.f;
  for (int s = wave; s < S_; s += 8) {
    float qd = q[(h * S_ + s) * D_ + d];
    const float* ar = aw + (h * S_ + s) * S_;
    const float* pr = p + (h * S_ + s) * S_;
    float x[8], mx = -3.402823466e38f;
    for (int i = 0; i < 8; ++i) {
      x[i] = (ar[lane + 32 * i] - qd * kd[i]) * SCALE;
      mx = fmaxf(mx, x[i]);
    }
    for (int o = 16; o >= 1; o >>= 1) mx = fmaxf(mx, __shfl_xor(mx, o, 32));
    float sum = 0.f;
    for (int i = 0; i < 8; ++i) { x[i] = __expf(x[i] - mx); sum += x[i]; }
    for (int o = 16; o >= 1; o >>= 1) sum += __shfl_xor(sum, o, 32);
    float inv = 1.f / sum;
    for (int i = 0; i < 8; ++i) {
      float dd = x[i] * inv - pr[lane + 32 * i];
      acc += dd * dd;
    }
  }
  for (int o = 16; o >= 1; o >>= 1) acc += __shfl_xor(acc, o, 32);
  __shared__ float red[8];
  if (lane == 0) red[wave] = acc;
  __syncthreads();
  if (threadIdx.x == 0) {
    float t = 0.f;
    for (int w = 0; w < 8; ++w) t += red[w];
    qk_imp[h * 64 + d] = t;
  }
}

// ---------------------------------------------------------------------------
// Kernel 6: attn_out = p @ v per head (GQA: v head = h/2), K=256. WMMA.
__global__ __launch_bounds__(32) void pv_gemm(
    const float* __restrict__ p, const float* __restrict__ v,
    float* __restrict__ ao) {
  int blk = blockIdx.x;        // h*64 + tm*4 + tn
  int h = blk >> 6;
  int tm = (blk >> 2) & 15, tn = blk & 3;
  int lane = threadIdx.x, half = lane >> 4, l = lane & 15;
  int m0 = tm * 16, n0 = tn * 16;
  const float* A = p + h * S_ * S_;        // [s][t]
  const float* B = v + (h >> 1) * S_ * D_; // [t][d] row-major KxN
  v8f acc = {};
  const float* ap = A + (m0 + l) * S_ + 2 * half;
  const float* bp = B + (2 * half) * D_ + n0 + l;
  for (int kk = 0; kk < S_; kk += 4) {
    v2f a = { ap[kk], ap[kk + 1] };
    v2f b = { bp[kk * D_], bp[kk * D_ + D_] };
    acc = wmma_k4(a, b, acc);
  }
  for (int r = 0; r < 8; ++r)
    ao[(m0 + r + 8 * half) * HID_ + h * D_ + n0 + l] = acc[r];
}

// ---------------------------------------------------------------------------
// Kernel 7: v_importance / o_importance = column abs-sums of attn_out.
__global__ __launch_bounds__(256) void vimp_kernel(
    const float* __restrict__ ao, float* __restrict__ vimp,
    float* __restrict__ oimp) {
  int c = blockIdx.x * blockDim.x + threadIdx.x; // 0..1023
  float s = 0.f;
  for (int r = 0; r < S_; ++r) s += fabsf(ao[r * HID_ + c]);
  vimp[c] = s;
  oimp[c] = s;
}

// ---------------------------------------------------------------------------
// Kernel 8: out = attn_out(256x1024) @ Wo(1024x1024). WMMA, K=1024.
__global__ __launch_bounds__(32) void out_gemm(
    const float* __restrict__ ao, const float* __restrict__ Wo,
    float* __restrict__ out) {
  int blk = blockIdx.x;     // tm*64 + tn
  int tm = blk >> 6, tn = blk & 63;
  int lane = threadIdx.x, half = lane >> 4, l = lane & 15;
  int m0 = tm * 16, n0 = tn * 16;
  v8f acc = {};
  const float* ap = ao + (m0 + l) * HID_ + 2 * half;
  const float* bp = Wo + (2 * half) * HID_ + n0 + l;
  for (int kk = 0; kk < HID_; kk += 4) {
    v2f a = { ap[kk], ap[kk + 1] };
    v2f b = { bp[kk * HID_], bp[kk * HID_ + HID_] };
    acc = wmma_k4(a, b, acc);
  }
  for (int r = 0; r < 8; ++r)
    out[(m0 + r + 8 * half) * HID_ + n0 + l] = acc[r];
}

// ---------------------------------------------------------------------------
extern "C" void kernel_launch(void* const* d_in, const int* in_sizes, int n_in,
                              void* d_out, int out_size, void* d_ws,
                              size_t ws_size, hipStream_t stream) {
  const float* hs   = (const float*)d_in[0];
  const float* cosb = (const float*)d_in[1];
  const float* sinb = (const float*)d_in[2];
  const float* mask = (const float*)d_in[3];
  const float* Wq   = (const float*)d_in[4];
  const float* Wk   = (const float*)d_in[5];
  const float* Wv   = (const float*)d_in[6];
  const float* Wo   = (const float*)d_in[7];

  // Outputs, concatenated flat in reference return order.
  float* out   = (float*)d_out;          // 262144
  float* p     = out + 262144;           // 1048576
  float* qkimp = out + 1310720;          // 1024
  float* vimp  = out + 1311744;          // 1024
  float* oimp  = out + 1312768;          // 1024

  // Workspace: ~7.0 MB of f32.
  float* ws = (float*)d_ws;
  float* q  = ws;            // [H][S][D]   262144
  float* k  = ws + 262144;   // [KV][S][D]  131072
  float* v  = ws + 393216;   // [KV][S][D]  131072
  float* aw = ws + 524288;   // [H][S][S]   1048576 (pre-softmax, masked)
  float* ao = ws + 1572864;  // [S][H*D]    262144

  qkv_gemm     <<<2048, 32,  0, stream>>>(hs, Wq, Wk, Wv, q, k, v);
  rope_kernel  <<<768,  256, 0, stream>>>(q, k, cosb, sinb);
  scores_gemm  <<<4096, 32,  0, stream>>>(q, k, mask, aw);
  softmax_rows <<<512,  256, 0, stream>>>(aw, p);
  qk_imp_kernel<<<1024, 256, 0, stream>>>(q, k, aw, p, qkimp);
  pv_gemm      <<<1024, 32,  0, stream>>>(p, v, ao);
  vimp_kernel  <<<4,    256, 0, stream>>>(ao, vimp, oimp);
  out_gemm     <<<1024, 32,  0, stream>>>(ao, Wo, out);
}